// CTF_75488345194883
// MI455X (gfx1250) — compile-verified
//
#include <hip/hip_runtime.h>

// CTF least-squares on MI455X (gfx1250):
//  - global scalar normalization (EPS_LAMBDA=1 -> lam is the global max of |Y|^2)
//  - per-frequency G = A^H A via V_WMMA_F32_16X16X4_F32, A gathered implicitly from
//    zero-guard-banded interleaved LDS band rows (no boundary selects, b64 LDS loads)
//  - G lives in d_ws (129 * 648KB ~= 86MB, L2-resident on the 192MB L2)
//  - in-block complex Cholesky + forward/backward substitution, then band recombination

#define F_TOT 129
#define N_T   512
#define NK    32
#define NB    9
#define NCOL  288     // NB * NK
#define NTILE 18      // NCOL / 16
#define NPAIR 171     // NTILE*(NTILE+1)/2
#define PADV  1e-8f
#define GUARD 32      // front zero pad per band row (max lag = 31)

typedef __attribute__((ext_vector_type(2))) float v2f;
typedef __attribute__((ext_vector_type(8))) float v8f;

// Load one 16x4-f32 WMMA fragment pair (real+imag) for K = i0, i0+1.
// Rows are interleaved (re,im) float2 with a 32-element zero guard band in front,
// so no boundary handling is needed and each K element is one ds_load_b64.
__device__ __forceinline__ void load_frag2(const float2* row, int i0, v2f& ar, v2f& ai) {
  float2 p0 = row[i0];
  float2 p1 = row[i0 + 1];
  ar.x = p0.x; ai.x = p0.y;
  ar.y = p1.x; ai.y = p1.y;
}

__global__ void maxp_init_kernel(unsigned* m) {
  if (threadIdx.x == 0 && blockIdx.x == 0) *m = 0u;
}

__global__ __launch_bounds__(256) void maxp_reduce_kernel(
    const float* __restrict__ Yre, const float* __restrict__ Yim,
    unsigned* m, int n) {
  __shared__ float red[256];
  float p = 0.0f;
  for (int i = blockIdx.x * blockDim.x + threadIdx.x; i < n;
       i += gridDim.x * blockDim.x) {
    float a = Yre[i], b = Yim[i];
    p = fmaxf(p, a * a + b * b);
  }
  red[threadIdx.x] = p;
  __syncthreads();
  for (int s = 128; s > 0; s >>= 1) {
    if (threadIdx.x < s)
      red[threadIdx.x] = fmaxf(red[threadIdx.x], red[threadIdx.x + s]);
    __syncthreads();
  }
  if (threadIdx.x == 0) atomicMax(m, __float_as_uint(red[0]));  // p >= 0: uint order == float order
}

__global__ __launch_bounds__(256) void ctf_solve_kernel(
    const float* __restrict__ Yre, const float* __restrict__ Yim,
    const float* __restrict__ Sre, const float* __restrict__ Sim,
    const unsigned* __restrict__ maxp_bits,
    float* __restrict__ Gws, float* __restrict__ outp, int fbase) {
  const int f = fbase + blockIdx.x;
  float2* G = (float2*)(Gws + (size_t)blockIdx.x * (2u * NCOL * NCOL));

  __shared__ float2 xc[NB][GUARD + N_T];   // interleaved (re,im), 32-elem zero guard
  __shared__ float2 yc[N_T];
  __shared__ float2 bvec[NCOL];
  __shared__ float2 zvec[NCOL];
  __shared__ float2 colL[NCOL];
  __shared__ float sdiag;

  const int tid = threadIdx.x;
  const float rsl = rsqrtf(__uint_as_float(maxp_bits[0]));  // 1/sqrt(max p)

  // ---- Stage 9 guard-banded band rows (normalized, PAD_VAL at freq edges) + y ----
  for (int w = 0; w < NB; ++w) {
    int fw = f + w - 4;
    bool inb = (fw >= 0) && (fw < F_TOT);
    const float* sr = Sre + (size_t)(inb ? fw : 0) * N_T;
    const float* si = Sim + (size_t)(inb ? fw : 0) * N_T;
    for (int t = tid; t < GUARD + N_T; t += 256) {
      float2 v;
      if (t < GUARD) {
        v = make_float2(0.0f, 0.0f);
      } else if (inb) {
        v = make_float2(sr[t - GUARD] * rsl, si[t - GUARD] * rsl);
      } else {
        v = make_float2(PADV, 0.0f);
      }
      xc[w][t] = v;
    }
  }
  for (int t = tid; t < N_T; t += 256) {
    yc[t] = make_float2(Yre[(size_t)f * N_T + t] * rsl,
                        Yim[(size_t)f * N_T + t] * rsl);
  }
  __syncthreads();

  // ---- G = A^H A with f32 WMMA; Hermitian -> only tile pairs t1<=t2 ----
  const int wave = tid >> 5;
  const int lane = tid & 31;
  const int hlf  = lane >> 4;    // K-half (ISA 7.12.2 f32 A/B layout)
  const int r    = lane & 15;    // M row (A operand) / N col (B operand)

  for (int p = wave; p < NPAIR; p += 8) {
    int t1 = 0, acc = 0;
    while (acc + (NTILE - t1) <= p) { acc += NTILE - t1; ++t1; }
    int t2 = t1 + (p - acc);

    const int off1 = 2 * hlf - (((t1 & 1) << 4) + r);  // K base minus lag (>= -31)
    const int off2 = 2 * hlf - (((t2 & 1) << 4) + r);
    const float2* row1 = &xc[t1 >> 1][GUARD];
    const float2* row2 = &xc[t2 >> 1][GUARD];

    v8f rr = {}, ii = {}, ri = {}, ir = {};
#pragma unroll 4
    for (int m0 = 0; m0 < N_T; m0 += 4) {
      v2f ar1, ai1, ar2, ai2;
      load_frag2(row1, m0 + off1, ar1, ai1);
      load_frag2(row2, m0 + off2, ar2, ai2);
      rr = __builtin_amdgcn_wmma_f32_16x16x4_f32(false, ar1, false, ar2, (short)0, rr, false, false);
      ii = __builtin_amdgcn_wmma_f32_16x16x4_f32(false, ai1, false, ai2, (short)0, ii, false, false);
      ri = __builtin_amdgcn_wmma_f32_16x16x4_f32(false, ar1, false, ai2, (short)0, ri, false, false);
      ir = __builtin_amdgcn_wmma_f32_16x16x4_f32(false, ai1, false, ar2, (short)0, ir, false, false);
    }

    // G_re = Ar'Ar + Ai'Ai ; G_im = Ar'Ai - Ai'Ar ; mirror the Hermitian conjugate
    const int n1 = t1 * 16, n2 = t2 * 16;
    for (int v = 0; v < 8; ++v) {
      int i = v + 8 * hlf, j = r;
      float re = rr[v] + ii[v];
      float im = ri[v] - ir[v];
      if (t1 == t2) {
        if (i <= j) {
          G[(size_t)(n1 + i) * NCOL + (n2 + j)] = make_float2(re, im);
          if (i < j) G[(size_t)(n2 + j) * NCOL + (n1 + i)] = make_float2(re, -im);
        }
      } else {
        G[(size_t)(n1 + i) * NCOL + (n2 + j)] = make_float2(re, im);
        G[(size_t)(n2 + j) * NCOL + (n1 + i)] = make_float2(re, -im);
      }
    }
  }

  // ---- b = A^H y ----
  for (int n = tid; n < NCOL; n += 256) {
    int w = n >> 5, lag = n & 31;
    const float2* row = &xc[w][GUARD - lag];
    float br = 0.0f, bi = 0.0f;
    for (int m = lag; m < N_T; ++m) {
      float2 a = row[m];
      float2 y = yc[m];
      br += a.x * y.x + a.y * y.y;    // conj(a)*y
      bi += a.x * y.y - a.y * y.x;
    }
    bvec[n] = make_float2(br, bi);
  }
  __threadfence_block();
  __syncthreads();

  // ---- Complex Cholesky G = L L^H (lower triangle, in place, L2-resident) ----
  for (int j = 0; j < NCOL; ++j) {
    if (tid == 0) {
      float d = sqrtf(fmaxf(G[(size_t)j * NCOL + j].x, 1e-30f));
      G[(size_t)j * NCOL + j] = make_float2(d, 0.0f);
      sdiag = d;
    }
    __syncthreads();
    float inv = 1.0f / sdiag;
    for (int i2 = j + 1 + tid; i2 < NCOL; i2 += 256) {
      float2 L = G[(size_t)i2 * NCOL + j];
      L.x *= inv; L.y *= inv;
      G[(size_t)i2 * NCOL + j] = L;
      colL[i2] = L;                 // cache column j in LDS for the rank-1 update
    }
    __threadfence_block();
    __syncthreads();
    for (int i2 = j + 1 + tid; i2 < NCOL; i2 += 256) {
      float2 Li = colL[i2];
      for (int k2 = j + 1; k2 <= i2; ++k2) {
        float2 Lk = colL[k2];
        float2 g = G[(size_t)i2 * NCOL + k2];
        g.x -= Li.x * Lk.x + Li.y * Lk.y;   // g -= Li * conj(Lk)
        g.y -= Li.y * Lk.x - Li.x * Lk.y;
        G[(size_t)i2 * NCOL + k2] = g;
      }
    }
    __threadfence_block();
    __syncthreads();
  }

  // ---- Forward solve L z = b ----
  for (int j = 0; j < NCOL; ++j) {
    if (tid == 0) {
      float d = G[(size_t)j * NCOL + j].x;
      zvec[j] = make_float2(bvec[j].x / d, bvec[j].y / d);
    }
    __syncthreads();
    float2 zj = zvec[j];
    for (int i2 = j + 1 + tid; i2 < NCOL; i2 += 256) {
      float2 L = G[(size_t)i2 * NCOL + j];
      bvec[i2].x -= L.x * zj.x - L.y * zj.y;
      bvec[i2].y -= L.x * zj.y + L.y * zj.x;
    }
    __syncthreads();
  }

  // ---- Backward solve L^H x = z (x overwrites z) ----
  for (int j = NCOL - 1; j >= 0; --j) {
    if (tid == 0) {
      float d = G[(size_t)j * NCOL + j].x;
      zvec[j] = make_float2(zvec[j].x / d, zvec[j].y / d);
    }
    __syncthreads();
    float2 xj = zvec[j];
    for (int i2 = tid; i2 < j; i2 += 256) {
      float2 L = G[(size_t)j * NCOL + i2];   // (L^H)[i2][j] = conj(L[j][i2])
      zvec[i2].x -= L.x * xj.x + L.y * xj.y;
      zvec[i2].y -= L.x * xj.y - L.y * xj.x;
    }
    __syncthreads();
  }

  // ---- out[f,k] = sum_w x[32w+k] * mu_w, mu_w = (-1)^(f+w-4) (PAD_VAL at edges) ----
  if (tid < NK) {
    float2 accu = make_float2(0.0f, 0.0f);
    for (int w = 0; w < NB; ++w) {
      int fw = f + w - 4;
      float m = (fw >= 0 && fw < F_TOT) ? ((fw & 1) ? -1.0f : 1.0f) : PADV;
      float2 s = zvec[w * NK + tid];
      accu.x += s.x * m;
      accu.y += s.y * m;
    }
    ((float2*)outp)[(size_t)f * NK + tid] = accu;   // interleaved complex64
  }
}

extern "C" void kernel_launch(void* const* d_in, const int* in_sizes, int n_in,
                              void* d_out, int out_size, void* d_ws, size_t ws_size,
                              hipStream_t stream) {
  (void)in_sizes; (void)n_in; (void)out_size;
  const float* Yre = (const float*)d_in[0];
  const float* Yim = (const float*)d_in[1];
  const float* Sre = (const float*)d_in[2];
  const float* Sim = (const float*)d_in[3];

  unsigned* maxp = (unsigned*)d_ws;
  float* Gbase = (float*)((char*)d_ws + 256);
  const size_t slotBytes = (size_t)2 * NCOL * NCOL * sizeof(float);  // 663,552 B / freq

  size_t avail = (ws_size > 256) ? (ws_size - 256) : 0;
  int slots = (int)(avail / slotBytes);
  if (slots > F_TOT) slots = F_TOT;
  if (slots < 1) slots = 1;  // best effort

  maxp_init_kernel<<<1, 1, 0, stream>>>(maxp);
  maxp_reduce_kernel<<<(F_TOT * N_T + 255) / 256, 256, 0, stream>>>(Yre, Yim, maxp, F_TOT * N_T);

  for (int base = 0; base < F_TOT; base += slots) {
    int cnt = F_TOT - base;
    if (cnt > slots) cnt = slots;
    ctf_solve_kernel<<<cnt, 256, 0, stream>>>(Yre, Yim, Sre, Sim, maxp, Gbase,
                                              (float*)d_out, base);
  }
}